// FullLorentzGraphNeuralNetwork_21199958573130
// MI455X (gfx1250) — compile-verified
//
#include <hip/hip_runtime.h>
#include <math.h>

typedef __attribute__((ext_vector_type(2))) float v2f;
typedef __attribute__((ext_vector_type(8))) float v8f;

#define NROWS 12288
#define DIN   128
#define DOUT  64
#define EPSF       1e-7f
#define ACOSH_EPSF 1e-5f
#define KSPLIT 2

// LDS staging geometry for kernel 2 (B = hT chunk shared by all 8 waves)
#define CHUNK_K 96                 // K elements per chunk (6144 / 96 = 64 chunks)
#define CSTEPS  (CHUNK_K / 4)      // 24 WMMA K-steps per chunk
#define BSTRIDE 100                // 96 + 4 pad floats: bank step 36 -> conflict-free

typedef __attribute__((address_space(3))) float lds_float;

__device__ __forceinline__ float wave_sum32(float v) {
    v += __shfl_xor(v, 16, 32);
    v += __shfl_xor(v, 8, 32);
    v += __shfl_xor(v, 4, 32);
    v += __shfl_xor(v, 2, 32);
    v += __shfl_xor(v, 1, 32);
    return v;
}

// ---------------------------------------------------------------------------
// Kernel 1: u = logmap0(x); v = u @ W^T + b; h = expmap0(v)
// Writes h TRANSPOSED: hT[c * NROWS + row] (64 x 12288) so K2's WMMA
// B-operand (lane = column, K striped) is a contiguous b64 load.
// ---------------------------------------------------------------------------
__global__ __launch_bounds__(256) void k1_feature(const float* __restrict__ x,
                                                  const float* __restrict__ W,
                                                  const float* __restrict__ b,
                                                  float* __restrict__ hT) {
    __shared__ float wlds[63 * 127];
    __shared__ float blds[63];
    __shared__ float ulds[8][128];

    const int tid = threadIdx.x;
    for (int i = tid; i < 63 * 127; i += 256) wlds[i] = W[i];
    if (tid < 63) blds[tid] = b[tid];
    __syncthreads();

    const int wave = tid >> 5;
    const int lane = tid & 31;
    const int row  = blockIdx.x * 8 + wave;

    const float4 xv = ((const float4*)(x + (size_t)row * DIN))[lane];
    float s = xv.y * xv.y + xv.z * xv.z + xv.w * xv.w;
    if (lane != 0) s += xv.x * xv.x;            // lane 0's .x is the time component
    s = wave_sum32(s);
    const float norm  = fmaxf(sqrtf(s), EPSF);
    const float x0    = __shfl(xv.x, 0, 32);
    const float theta = acoshf(fmaxf(x0, 1.0f + ACOSH_EPSF));
    const float sc    = theta / norm;

    float* u = ulds[wave];
    if (lane != 0) u[4 * lane - 1] = sc * xv.x;
    u[4 * lane + 0] = sc * xv.y;
    u[4 * lane + 1] = sc * xv.z;
    u[4 * lane + 2] = sc * xv.w;
    __syncthreads();

    const int o1  = lane;
    const int o2  = lane + 32;
    const int o2e = (o2 < 63) ? o2 : 62;
    float acc1 = blds[o1];
    float acc2 = blds[o2e];
    const float* w1 = &wlds[o1 * 127];
    const float* w2 = &wlds[o2e * 127];
    #pragma unroll 4
    for (int i = 0; i < 127; ++i) {
        const float uv = u[i];
        acc1 = fmaf(uv, w1[i], acc1);
        acc2 = fmaf(uv, w2[i], acc2);
    }
    if (o2 >= 63) acc2 = 0.0f;

    const float nv2 = wave_sum32(acc1 * acc1 + acc2 * acc2);
    const float nrm = fmaxf(sqrtf(nv2), EPSF);
    const float h0  = coshf(nrm);
    const float ssc = sinhf(nrm) / nrm;

    hT[(size_t)(o1 + 1) * NROWS + row] = ssc * acc1;
    if (o2 < 63) hT[(size_t)(o2 + 1) * NROWS + row] = ssc * acc2;
    if (lane == 0) hT[row] = h0;
}

// ---------------------------------------------------------------------------
// Kernel 2: m = adj @ h via V_WMMA_F32_16X16X4_F32.
//  - A (adj) streamed from HBM with global_load_b64 + global_prefetch.
//  - B (hT) chunks staged into LDS with GLOBAL_LOAD_ASYNC_TO_LDS_B128
//    (ASYNCcnt), double-buffered so the async DMA overlaps WMMA compute;
//    read back as conflict-free ds_load_b64 (padded column stride 100).
// Each wave: 16(M) x 64(N) strip over one of KSPLIT K-chunks.
// ---------------------------------------------------------------------------
__global__ __launch_bounds__(256) void k2_spmm(const float* __restrict__ adj,
                                               const float* __restrict__ hT,
                                               float* __restrict__ mpart) {
    __shared__ float blds[2][64][BSTRIDE];       // 50 KB double buffer

    const int tid  = threadIdx.x;
    const int wave = tid >> 5;
    const int lane = tid & 31;

    // Whole block shares one K-half so LDS B-chunks are common to all 8 waves.
    const int ks   = blockIdx.x & (KSPLIT - 1);
    const int grp  = blockIdx.x >> 1;            // 0 .. 95
    const int rblk = grp * 8 + wave;             // 0 .. 767
    const int row0 = rblk * 16;
    const int k0   = ks * (NROWS / KSPLIT);

    const int mrow  = lane & 15;
    const int khalf = lane >> 4;

    const float* ap = adj + (size_t)(row0 + mrow) * NROWS + k0 + 2 * khalf;

    lds_float* lbase0 = (lds_float*)&blds[0][0][0];
    lds_float* lbase1 = (lds_float*)&blds[1][0][0];

    // Async fill of one 64-col x CHUNK_K B-chunk: 24576 B = 1536 x 16 B pieces,
    // 6 lane-wide b128 async copies per wave.
    auto fill_chunk = [&](int nb, int kbase) {
        lds_float* lb = nb ? lbase1 : lbase0;
        #pragma unroll
        for (int t = 0; t < 6; ++t) {
            const int piece = wave * 192 + t * 32 + lane;   // 0 .. 1535
            const int col   = piece / 24;                   // 24 x 16B per column
            const int k16   = piece - col * 24;
            const float* g  = hT + (size_t)col * NROWS + kbase + (k16 << 2);
            lds_float* l    = lb + col * BSTRIDE + (k16 << 2);
            asm volatile("global_load_async_to_lds_b128 %0, %1, off"
                         :: "v"(l), "v"(g) : "memory");
        }
    };

    fill_chunk(0, k0);
    asm volatile("s_wait_asynccnt 0x0" ::: "memory");
    __syncthreads();

    v8f c0 = {}, c1 = {}, c2 = {}, c3 = {};
    const int c0i = mrow, c1i = mrow + 16, c2i = mrow + 32, c3i = mrow + 48;
    const int nchunks = (NROWS / KSPLIT) / CHUNK_K;          // 64
    int buf = 0;

    for (int ch = 0; ch < nchunks; ++ch) {
        if (ch + 1 < nchunks) fill_chunk(buf ^ 1, k0 + (ch + 1) * CHUNK_K);

        #pragma unroll 4
        for (int s = 0; s < CSTEPS; ++s) {
            if ((s & 3) == 0) __builtin_prefetch(ap + 1024, 0, 0);  // adj +4KB
            const int kk = (s << 2) + (khalf << 1);
            const v2f a  = *(const v2f*)ap; ap += 4;
            const v2f b0 = *(const v2f*)&blds[buf][c0i][kk];
            const v2f b1 = *(const v2f*)&blds[buf][c1i][kk];
            const v2f b2 = *(const v2f*)&blds[buf][c2i][kk];
            const v2f b3 = *(const v2f*)&blds[buf][c3i][kk];
            c0 = __builtin_amdgcn_wmma_f32_16x16x4_f32(false, a, false, b0, (short)0, c0, false, false);
            c1 = __builtin_amdgcn_wmma_f32_16x16x4_f32(false, a, false, b1, (short)0, c1, false, false);
            c2 = __builtin_amdgcn_wmma_f32_16x16x4_f32(false, a, false, b2, (short)0, c2, false, false);
            c3 = __builtin_amdgcn_wmma_f32_16x16x4_f32(false, a, false, b3, (short)0, c3, false, false);
        }

        asm volatile("s_wait_asynccnt 0x0" ::: "memory");
        __syncthreads();
        buf ^= 1;
    }

    // C/D layout: VGPR v -> row (row0 + v + 8*khalf), lane&15 -> column in tile
    float* mp = mpart + (size_t)ks * ((size_t)NROWS * DOUT);
    const int rowoff = row0 + 8 * khalf;
    #pragma unroll
    for (int v = 0; v < 8; ++v) {
        const size_t rbase = (size_t)(rowoff + v) * DOUT + mrow;
        mp[rbase +  0] = c0[v];
        mp[rbase + 16] = c1[v];
        mp[rbase + 32] = c2[v];
        mp[rbase + 48] = c3[v];
    }
}

// ---------------------------------------------------------------------------
// Kernel 3: sum K-partials, Lorentz-normalize, logmap0 -> relu -> expmap0.
// One wave per row; lane L handles columns 2L, 2L+1.
// ---------------------------------------------------------------------------
__global__ __launch_bounds__(256) void k3_post(const float* __restrict__ mpart,
                                               float* __restrict__ out) {
    const int tid  = threadIdx.x;
    const int wave = tid >> 5;
    const int lane = tid & 31;
    const int row  = blockIdx.x * 8 + wave;

    const float2 a  = ((const float2*)(mpart + (size_t)row * DOUT))[lane];
    const float2 bb = ((const float2*)(mpart + (size_t)(NROWS + row) * DOUT))[lane];
    const float e0 = a.x + bb.x;
    const float e1 = a.y + bb.y;

    const float S  = wave_sum32(e0 * e0 + e1 * e1);
    const float m0 = __shfl(e0, 0, 32);
    const float inner = S - 2.0f * m0 * m0;             // -m0^2 + sum_{c>=1}
    const float inv   = 1.0f / sqrtf(fmaxf(fabsf(inner), EPSF));

    const float hr0 = m0 * inv;
    const float he0 = e0 * inv;
    const float he1 = e1 * inv;

    const float ns2    = fmaxf((S - m0 * m0) * inv * inv, 0.0f);
    const float norm_s = fmaxf(sqrtf(ns2), EPSF);
    const float theta  = acoshf(fmaxf(hr0, 1.0f + ACOSH_EPSF));
    const float sc     = theta / norm_s;

    const float u0 = (lane == 0) ? 0.0f : fmaxf(sc * he0, 0.0f);
    const float u1 = fmaxf(sc * he1, 0.0f);

    const float nr2 = wave_sum32(u0 * u0 + u1 * u1);
    const float nrm = fmaxf(sqrtf(nr2), EPSF);
    const float o0  = coshf(nrm);
    const float ssc = sinhf(nrm) / nrm;

    float2 r;
    r.x = (lane == 0) ? o0 : ssc * u0;
    r.y = ssc * u1;
    ((float2*)(out + (size_t)row * DOUT))[lane] = r;
}

// ---------------------------------------------------------------------------
extern "C" void kernel_launch(void* const* d_in, const int* in_sizes, int n_in,
                              void* d_out, int out_size, void* d_ws, size_t ws_size,
                              hipStream_t stream) {
    const float* x   = (const float*)d_in[0];   // (12288, 128)
    const float* adj = (const float*)d_in[1];   // (12288, 12288)
    const float* W   = (const float*)d_in[2];   // (63, 127)
    const float* b   = (const float*)d_in[3];   // (63,)
    float* out = (float*)d_out;                 // (12288, 64)

    float* hT    = (float*)d_ws;                        // 64 x 12288 floats (3 MB)
    float* mpart = hT + (size_t)DOUT * NROWS;           // KSPLIT x (12288 x 64) (6 MB)

    hipLaunchKernelGGL(k1_feature, dim3(NROWS / 8), dim3(256), 0, stream, x, W, b, hT);
    hipLaunchKernelGGL(k2_spmm, dim3((NROWS / 16) * KSPLIT / 8), dim3(256), 0, stream,
                       adj, hT, mpart);
    hipLaunchKernelGGL(k3_post, dim3(NROWS / 8), dim3(256), 0, stream, mpart, out);
}